// GIN_46445776339725
// MI455X (gfx1250) — compile-verified
//
#include <hip/hip_runtime.h>

typedef float v2f __attribute__((ext_vector_type(2)));
typedef float v8f __attribute__((ext_vector_type(8)));

#define NN 100000
#define NE 3200000
#define NG 2048
#define DH 32
#define D3 96
#define NC 10
#define SLOPE 0.01f
#define BN_EPS 1e-5f

// ---------------- zero ----------------
__global__ void zero_kernel(float* __restrict__ p, int n) {
    int i = blockIdx.x * blockDim.x + threadIdx.x;
    int stride = gridDim.x * blockDim.x;
    for (; i < n; i += stride) p[i] = 0.0f;
}

// ---------------- edge scatter-add: wave per edge, lane per feature ----------------
// agg (12.8MB) + h (12.8MB) are L2-resident; hardware f32 atomics keep the RMW in L2.
__global__ void scatter_kernel(const float* __restrict__ h,
                               const long long* __restrict__ ei,
                               float* __restrict__ agg) {
    long long tid = (long long)blockIdx.x * blockDim.x + threadIdx.x;
    int lane = threadIdx.x & 31;
    long long e = tid >> 5;
    if (e >= NE) return;
    int src = (int)ei[e];
    int dst = (int)ei[(long long)NE + e];
    unsafeAtomicAdd(&agg[(long long)dst * DH + lane], h[(long long)src * DH + lane]);
}

// ---------------- fused GIN MLP: (h+agg)@w1^T+b1 -> BN -> lrelu -> @w2^T+b2 -> lrelu ----------------
__global__ void __launch_bounds__(256)
gin_mlp_kernel(const float* __restrict__ h_in, const float* __restrict__ agg,
               const float* __restrict__ w1, const float* __restrict__ b1,
               const float* __restrict__ g,  const float* __restrict__ bt,
               const float* __restrict__ bm, const float* __restrict__ bv,
               const float* __restrict__ w2, const float* __restrict__ b2,
               float* __restrict__ h_out) {
    __shared__ float lds[8][16 * DH];
    int lane = threadIdx.x & 31;
    int wv   = threadIdx.x >> 5;
    int tile = blockIdx.x * 8 + wv;
    if (tile * 16 >= NN) return;            // wave-uniform: EXEC stays all-1s for WMMA
    float* Z = lds[wv];
    int row0 = tile * 16;

    // coalesced load of z = h + agg (16 rows x 32 cols) into LDS
    for (int r = 0; r < 16; ++r) {
        long long idx = (long long)(row0 + r) * DH + lane;
        Z[r * DH + lane] = h_in[idx] + agg[idx];
    }
    asm volatile("s_wait_dscnt 0" ::: "memory");

    int mn = lane & 15;            // A: row M ; B: col N ; C/D: col N
    int kg = (lane >> 4) * 2;      // K sub-offset within 16x4 fragment

    // ---- GEMM1: z @ w1^T ----
    v8f acc0 = {}; v8f acc1 = {};
    for (int k0 = 0; k0 < DH; k0 += 4) {
        int kk = k0 + kg;
        v2f a;  a.x  = Z[mn * DH + kk];          a.y  = Z[mn * DH + kk + 1];
        v2f bl; bl.x = w1[mn * DH + kk];         bl.y = w1[mn * DH + kk + 1];
        v2f bh; bh.x = w1[(mn + 16) * DH + kk];  bh.y = w1[(mn + 16) * DH + kk + 1];
        acc0 = __builtin_amdgcn_wmma_f32_16x16x4_f32(false, a, false, bl, (short)0, acc0, false, false);
        acc1 = __builtin_amdgcn_wmma_f32_16x16x4_f32(false, a, false, bh, (short)0, acc1, false, false);
    }

    // ---- epilogue1: bias + BN(eval) + leakyrelu, write back to LDS in row-major ----
    {
        int n1 = mn, n2 = mn + 16;
        float s1 = g[n1] * rsqrtf(bv[n1] + BN_EPS);
        float t1 = (b1[n1] - bm[n1]) * s1 + bt[n1];
        float s2 = g[n2] * rsqrtf(bv[n2] + BN_EPS);
        float t2 = (b1[n2] - bm[n2]) * s2 + bt[n2];
        int mbase = (lane >> 4) * 8;
        #pragma unroll
        for (int j = 0; j < 8; ++j) {
            float t = acc0[j] * s1 + t1;  t = t > 0.0f ? t : t * SLOPE;
            Z[(mbase + j) * DH + n1] = t;
            float u = acc1[j] * s2 + t2;  u = u > 0.0f ? u : u * SLOPE;
            Z[(mbase + j) * DH + n2] = u;
        }
    }
    asm volatile("s_wait_dscnt 0" ::: "memory");

    // ---- GEMM2: z1 @ w2^T ----
    v8f d0 = {}; v8f d1 = {};
    for (int k0 = 0; k0 < DH; k0 += 4) {
        int kk = k0 + kg;
        v2f a;  a.x  = Z[mn * DH + kk];          a.y  = Z[mn * DH + kk + 1];
        v2f bl; bl.x = w2[mn * DH + kk];         bl.y = w2[mn * DH + kk + 1];
        v2f bh; bh.x = w2[(mn + 16) * DH + kk];  bh.y = w2[(mn + 16) * DH + kk + 1];
        d0 = __builtin_amdgcn_wmma_f32_16x16x4_f32(false, a, false, bl, (short)0, d0, false, false);
        d1 = __builtin_amdgcn_wmma_f32_16x16x4_f32(false, a, false, bh, (short)0, d1, false, false);
    }

    // ---- epilogue2: bias + leakyrelu, store ----
    {
        float c1 = b2[mn], c2 = b2[mn + 16];
        int mbase = (lane >> 4) * 8;
        #pragma unroll
        for (int j = 0; j < 8; ++j) {
            float t = d0[j] + c1;  t = t > 0.0f ? t : t * SLOPE;
            h_out[(long long)(row0 + mbase + j) * DH + mn] = t;
            float u = d1[j] + c2;  u = u > 0.0f ? u : u * SLOPE;
            h_out[(long long)(row0 + mbase + j) * DH + mn + 16] = u;
        }
    }
}

// ---------------- graph pooling: wave per node, lane per feature ----------------
__global__ void pool_kernel(const float* __restrict__ h,
                            const long long* __restrict__ batch,
                            float* __restrict__ pool, int off) {
    long long tid = (long long)blockIdx.x * blockDim.x + threadIdx.x;
    int lane = threadIdx.x & 31;
    long long node = tid >> 5;
    if (node >= NN) return;
    int gidx = (int)batch[node];
    unsafeAtomicAdd(&pool[(long long)gidx * D3 + off + lane], h[node * DH + lane]);
}

// ---------------- readout l1: pool[2048,96] @ l1_w^T[96,96] + b, relu ----------------
__global__ void __launch_bounds__(256)
mlp1_kernel(const float* __restrict__ pool, const float* __restrict__ w,
            const float* __restrict__ b, float* __restrict__ hid) {
    int lane = threadIdx.x & 31;
    int wv   = threadIdx.x >> 5;
    int wid  = blockIdx.x * 8 + wv;          // 0..383 : 128 row tiles x 3 column pairs
    if (wid >= (NG / 16) * 3) return;
    int rowTile = wid / 3, pair = wid % 3;
    int row0 = rowTile * 16;
    int n0 = pair * 32;
    int mn = lane & 15, kg = (lane >> 4) * 2;

    v8f acc0 = {}; v8f acc1 = {};
    for (int k0 = 0; k0 < D3; k0 += 4) {
        int kk = k0 + kg;
        v2f a;  a.x  = pool[(long long)(row0 + mn) * D3 + kk];
                a.y  = pool[(long long)(row0 + mn) * D3 + kk + 1];
        v2f bl; bl.x = w[(n0 + mn) * D3 + kk];       bl.y = w[(n0 + mn) * D3 + kk + 1];
        v2f bh; bh.x = w[(n0 + 16 + mn) * D3 + kk];  bh.y = w[(n0 + 16 + mn) * D3 + kk + 1];
        acc0 = __builtin_amdgcn_wmma_f32_16x16x4_f32(false, a, false, bl, (short)0, acc0, false, false);
        acc1 = __builtin_amdgcn_wmma_f32_16x16x4_f32(false, a, false, bh, (short)0, acc1, false, false);
    }
    float bb0 = b[n0 + mn], bb1 = b[n0 + 16 + mn];
    int mbase = (lane >> 4) * 8;
    #pragma unroll
    for (int j = 0; j < 8; ++j) {
        float t = acc0[j] + bb0;  t = fmaxf(t, 0.0f);
        hid[(long long)(row0 + mbase + j) * D3 + n0 + mn] = t;
        float u = acc1[j] + bb1;  u = fmaxf(u, 0.0f);
        hid[(long long)(row0 + mbase + j) * D3 + n0 + 16 + mn] = u;
    }
}

// ---------------- readout l2: hid[2048,96] @ l2_w^T[96,10] + b, leakyrelu ----------------
__global__ void mlp2_kernel(const float* __restrict__ hid, const float* __restrict__ w,
                            const float* __restrict__ b, float* __restrict__ out) {
    int tid = blockIdx.x * blockDim.x + threadIdx.x;
    if (tid >= NG * NC) return;
    int gidx = tid / NC, c = tid % NC;
    float acc = b[c];
    for (int k = 0; k < D3; ++k) acc += hid[(long long)gidx * D3 + k] * w[c * D3 + k];
    out[tid] = acc > 0.0f ? acc : acc * SLOPE;
}

extern "C" void kernel_launch(void* const* d_in, const int* in_sizes, int n_in,
                              void* d_out, int out_size, void* d_ws, size_t ws_size,
                              hipStream_t stream) {
    const float*     x     = (const float*)d_in[0];
    // d_in[1] = x_e (unused by reference)
    const long long* ei    = (const long long*)d_in[2];
    const long long* batch = (const long long*)d_in[3];
    const float* c[3][8];
    for (int l = 0; l < 3; ++l)
        for (int p = 0; p < 8; ++p)
            c[l][p] = (const float*)d_in[4 + l * 8 + p];
    const float* l1w = (const float*)d_in[28];
    const float* l1b = (const float*)d_in[29];
    const float* l2w = (const float*)d_in[30];
    const float* l2b = (const float*)d_in[31];
    float* out = (float*)d_out;

    char* ws = (char*)d_ws;
    size_t hbytes = (size_t)NN * DH * sizeof(float);
    float* agg  = (float*)(ws);
    float* bufA = (float*)(ws + hbytes);
    float* bufB = (float*)(ws + 2 * hbytes);
    float* pool = (float*)(ws + 3 * hbytes);
    float* hid  = (float*)(ws + 3 * hbytes + (size_t)NG * D3 * sizeof(float));

    dim3 blk(256);
    int zeroBlocksH   = (NN * DH + 255) / 256;
    int zeroBlocksP   = (NG * D3 + 255) / 256;
    int scatterBlocks = (int)(((long long)NE * 32 + 255) / 256);
    int ginBlocks     = ((NN / 16) + 7) / 8;
    int poolBlocks    = (int)(((long long)NN * 32 + 255) / 256);
    int l1Blocks      = ((NG / 16) * 3 + 7) / 8;
    int l2Blocks      = (NG * NC + 255) / 256;

    zero_kernel<<<zeroBlocksP, blk, 0, stream>>>(pool, NG * D3);

    const float* hcur = x;
    float* houts[3] = { bufA, bufB, bufA };   // h1 dead once layer-2 consumed it
    for (int l = 0; l < 3; ++l) {
        zero_kernel<<<zeroBlocksH, blk, 0, stream>>>(agg, NN * DH);
        scatter_kernel<<<scatterBlocks, blk, 0, stream>>>(hcur, ei, agg);
        gin_mlp_kernel<<<ginBlocks, blk, 0, stream>>>(hcur, agg,
            c[l][0], c[l][1], c[l][2], c[l][3], c[l][4], c[l][5], c[l][6], c[l][7],
            houts[l]);
        pool_kernel<<<poolBlocks, blk, 0, stream>>>(houts[l], batch, pool, l * DH);
        hcur = houts[l];
    }
    mlp1_kernel<<<l1Blocks, blk, 0, stream>>>(pool, l1w, l1b, hid);
    mlp2_kernel<<<l2Blocks, blk, 0, stream>>>(hid, l2w, l2b, out);
}